// JacobiMachine_13039520710825
// MI455X (gfx1250) — compile-verified
//
#include <hip/hip_runtime.h>
#include <hip/hip_bf16.h>

// ----------------------------------------------------------------------------
// 512x512 Jacobi (cross stencil, 0.25 * 4-neighbor sum, boundary masked), 1000
// steps. Persistent-kernel design for MI455X:
//   * 16 workgroups (one cluster's worth), each owns a 32-row slab, LDS
//     resident, double buffered (2 x 34 x 516 f32 = 140,352 B < 320 KB/WGP).
//   * per-step cross-WG traffic = halo rows only, via a small L2 buffer.
//   * per-step sync = s_cluster_barrier (CDNA5; NOP if not clustered) backed
//     by an always-correct monotonic global atomic barrier.
//   * stencil computed as 0.25*(S*T + T*S) per 16x16 tile with
//     V_WMMA_F32_16X16X4_F32 (8 WMMAs/tile, accumulator-chained), halo
//     row/col contributions patched with VALU selects, mask applied on store.
// ----------------------------------------------------------------------------

#define HW      512
#define NT      1000
#define NWG     16
#define ROWS    32            // owned rows per WG
#define LROWS   34            // + top/bottom halo rows
#define LSTRIDE 516           // 512 cols + left/right halo col + 2 pad
#define TPB     256           // 8 waves of 32

typedef __attribute__((ext_vector_type(2))) float v2f;
typedef __attribute__((ext_vector_type(8))) float v8f;

__global__ void __launch_bounds__(TPB)
jacobi_cluster_wmma(const float* __restrict__ X, const float* __restrict__ Y,
                    float* __restrict__ out,
                    unsigned* __restrict__ bar_cnt,
                    float* __restrict__ halo /* [2 parity][2 top/bot][NWG][512] */)
{
    extern __shared__ float lds[];
    float* buf0 = lds;
    float* buf1 = lds + LROWS * LSTRIDE;

    const int wg   = blockIdx.x;
    const int tid  = threadIdx.x;
    const int wave = tid >> 5;
    const int lane = tid & 31;
    const int lo   = lane & 15;     // position within 16-lane half
    const int hi   = lane >> 4;     // which 16-lane half

    // ---- zero both buffers (establishes zero halo columns / pad forever) ----
    for (int i = tid; i < LROWS * LSTRIDE; i += TPB) { buf0[i] = 0.0f; buf1[i] = 0.0f; }
    __syncthreads();

    // ---- initial condition: x0 = exp(-50*((X-.5)^2+(Y-.5)^2)), UNmasked ----
    // LDS row r <-> global row wg*32 + r - 1 ; LDS col c <-> global col c - 1
    for (int i = tid; i < LROWS * HW; i += TPB) {
        int r = i >> 9;            // 0..33
        int c = i & 511;           // global col 0..511
        int g = wg * ROWS + r - 1; // global row
        float v = 0.0f;
        if (g >= 0 && g < HW) {
            float dx = X[g * HW + c] - 0.5f;
            float dy = Y[g * HW + c] - 0.5f;
            v = expf(-50.0f * (dx * dx + dy * dy));
        }
        buf0[r * LSTRIDE + (c + 1)] = v;
    }
    __syncthreads();

    // ---- per-lane constant slices of the shift-sum matrix S (0/1 off-diag) --
    // A-layout (16x4 f32): VGPR j holds K = 4k + j + 2*hi for row M = lo.
    // The same values serve as B-layout slices for the T*S pass (S symmetric).
    float ca[4][2];
#pragma unroll
    for (int k = 0; k < 4; ++k)
#pragma unroll
        for (int j = 0; j < 2; ++j) {
            int kk = 4 * k + j + 2 * hi;
            int d  = lo - kk;
            ca[k][j] = (d == 1 || d == -1) ? 1.0f : 0.0f;
        }

    // ---- time loop --------------------------------------------------------
    for (int t = 0; t < NT; ++t) {
        const float* cur = (t & 1) ? buf1 : buf0;
        float*       nxt = (t & 1) ? buf0 : buf1;

        // 64 tiles (2 tile-rows x 32 tile-cols); wave w -> tiles w, w+8, ...
        for (int tile = wave; tile < 64; tile += 8) {
            int tr = tile >> 5, tc = tile & 31;
            int R0 = 1 + 16 * tr;      // LDS row of tile origin
            int C0 = 1 + 16 * tc;      // LDS col of tile origin
            v8f acc = {};

            // vertical: acc += S(16x16) * T(16x16), K split into 4 chunks of 4
#pragma unroll
            for (int k = 0; k < 4; ++k) {
                v2f a = { ca[k][0], ca[k][1] };
                v2f b = { cur[(R0 + 4 * k + 2 * hi + 0) * LSTRIDE + C0 + lo],
                          cur[(R0 + 4 * k + 2 * hi + 1) * LSTRIDE + C0 + lo] };
                acc = __builtin_amdgcn_wmma_f32_16x16x4_f32(
                        false, a, false, b, (short)0, acc, false, false);
            }
            // horizontal: acc += T(16x16) * S(16x16)
#pragma unroll
            for (int k = 0; k < 4; ++k) {
                v2f a = { cur[(R0 + lo) * LSTRIDE + C0 + 4 * k + 2 * hi + 0],
                          cur[(R0 + lo) * LSTRIDE + C0 + 4 * k + 2 * hi + 1] };
                v2f b = { ca[k][0], ca[k][1] };
                acc = __builtin_amdgcn_wmma_f32_16x16x4_f32(
                        false, a, false, b, (short)0, acc, false, false);
            }

            // cross-tile vertical fixups: row 0 += T[-1], row 15 += T[16]
            // C/D layout: VGPR r = rows {r (lanes 0-15), r+8 (lanes 16-31)}
            float top = cur[(R0 - 1)  * LSTRIDE + C0 + lo];
            float bot = cur[(R0 + 16) * LSTRIDE + C0 + lo];
            acc[0] += hi ? 0.0f : top;
            acc[7] += hi ? bot  : 0.0f;

            // cross-tile horizontal fixups + scale + boundary mask + store
#pragma unroll
            for (int r = 0; r < 8; ++r) {
                int   m   = r + 8 * hi;                    // row within tile
                float lft = cur[(R0 + m) * LSTRIDE + C0 - 1];
                float rgt = cur[(R0 + m) * LSTRIDE + C0 + 16];
                float v   = acc[r];
                v += (lo == 0  ? lft : 0.0f);
                v += (lo == 15 ? rgt : 0.0f);
                v *= 0.25f;
                int g  = wg * ROWS + (R0 + m) - 1;         // global row
                int gc = (C0 + lo) - 1;                    // global col
                if (g == 0 || g == HW - 1 || gc == 0 || gc == HW - 1) v = 0.0f;
                nxt[(R0 + m) * LSTRIDE + C0 + lo] = v;
            }
        }
        __syncthreads();

        // ---- export this slab's boundary rows to the L2 halo buffer --------
        int par = t & 1;
        float* haloT = halo + ((size_t)(par * 2 + 0) * NWG + wg) * HW;
        float* haloB = halo + ((size_t)(par * 2 + 1) * NWG + wg) * HW;
        for (int c = tid; c < HW; c += TPB) {
            haloT[c] = nxt[1  * LSTRIDE + c + 1];
            haloB[c] = nxt[32 * LSTRIDE + c + 1];
        }
        __threadfence();            // release halo stores to device scope
        __syncthreads();

        // ---- cross-WG barrier ---------------------------------------------
        // CDNA5 cluster split-barrier: real sync when dispatched as a
        // 16-WG cluster; architected S_NOP otherwise.
        __builtin_amdgcn_s_cluster_barrier();
        // Always-correct fallback: monotonic agent-scope counter (zeroed by a
        // hipMemsetAsync node each launch, so graph replays stay correct).
        if (tid == 0) {
            __hip_atomic_fetch_add(bar_cnt, 1u, __ATOMIC_ACQ_REL,
                                   __HIP_MEMORY_SCOPE_AGENT);
            unsigned target = (unsigned)(t + 1) * NWG;
            while (__hip_atomic_load(bar_cnt, __ATOMIC_ACQUIRE,
                                     __HIP_MEMORY_SCOPE_AGENT) < target)
                __builtin_amdgcn_s_sleep(1);
        }
        __syncthreads();
        __threadfence();            // acquire: invalidate WGP-level caches

        // ---- import neighbor halo rows into nxt ----------------------------
        if (wg > 0) {
            const float* up = halo + ((size_t)(par * 2 + 1) * NWG + (wg - 1)) * HW;
            for (int c = tid; c < HW; c += TPB)
                nxt[0 * LSTRIDE + c + 1] = up[c];       // bottom row of wg-1
        }
        if (wg < NWG - 1) {
            const float* dn = halo + ((size_t)(par * 2 + 0) * NWG + (wg + 1)) * HW;
            for (int c = tid; c < HW; c += TPB)
                nxt[33 * LSTRIDE + c + 1] = dn[c];      // top row of wg+1
        }
        __syncthreads();
    }

    // ---- write final state (NT even -> result sits in buf0) ----------------
    const float* fin = (NT & 1) ? buf1 : buf0;
    for (int i = tid; i < ROWS * HW; i += TPB) {
        int r = i >> 9, c = i & 511;
        out[((size_t)(wg * ROWS + r)) * HW + c] = fin[(r + 1) * LSTRIDE + c + 1];
    }
}

extern "C" void kernel_launch(void* const* d_in, const int* in_sizes, int n_in,
                              void* d_out, int out_size, void* d_ws, size_t ws_size,
                              hipStream_t stream) {
    (void)in_sizes; (void)n_in; (void)out_size; (void)ws_size;
    const float* X = (const float*)d_in[0];
    const float* Y = (const float*)d_in[1];
    float* out = (float*)d_out;

    // workspace layout: [0,256) barrier counter (zeroed every call, so graph
    // replays are deterministic); [256, 256 + 2*2*NWG*512*4) halo buffer.
    unsigned* bar_cnt = (unsigned*)d_ws;
    float*    halo    = (float*)((char*)d_ws + 256);
    hipMemsetAsync(d_ws, 0, 256, stream);

    const size_t smem = (size_t)2 * LROWS * LSTRIDE * sizeof(float); // 140,352 B
    hipFuncSetAttribute((const void*)jacobi_cluster_wmma,
                        hipFuncAttributeMaxDynamicSharedMemorySize, (int)smem);

    jacobi_cluster_wmma<<<dim3(NWG), dim3(TPB), smem, stream>>>(
        X, Y, out, bar_cnt, halo);
}